// CorticalInteractionModule_60129542387
// MI455X (gfx1250) — compile-verified
//
#include <hip/hip_runtime.h>
#include <hip/hip_bf16.h>

#define B_ 8
#define T_ 2048
#define C_ 768
#define M_ (B_ * T_)      // 16384 rows
#define NTILES (C_ / 16)  // 48
#define KCHUNK 256
#define LPITCH 17         // LDS pitch (floats) per 16-wide row -> bank-conflict free
#define MSUB 4            // 4 x 16-row subtiles per wave -> 64x16 wave tile

typedef __attribute__((ext_vector_type(2))) float v2f;
typedef __attribute__((ext_vector_type(8))) float v8f;

// ---------------------------------------------------------------------------
// Fused projections: k = x*Wk^T, v = x*Wv^T, sr = sigmoid(x*Wr^T)
// One wave computes a 64x16 output tile for all three matrices (A fragments
// shared across matrices; B fragments shared across the 4 M-subtiles).
// Weight tiles staged in LDS per 256-wide K chunk, shared by all 8 waves.
// ---------------------------------------------------------------------------
__global__ __launch_bounds__(256) void proj_kernel(
    const float* __restrict__ x, const float* __restrict__ Wk,
    const float* __restrict__ Wv, const float* __restrict__ Wr,
    float* __restrict__ k, float* __restrict__ v, float* __restrict__ sr)
{
    __shared__ float ldsW[3 * KCHUNK * LPITCH];   // [mat][kk][n], padded pitch

    const int nTile  = blockIdx.x % NTILES;
    const int mBlock = blockIdx.x / NTILES;
    const int wave   = threadIdx.x >> 5;
    const int lane   = threadIdx.x & 31;
    const int mBase  = mBlock * (8 * 16 * MSUB) + wave * (16 * MSUB);
    const int nBase  = nTile * 16;

    v8f accK[MSUB] = {};
    v8f accV[MSUB] = {};
    v8f accR[MSUB] = {};

    const int rowOff = lane & 15;          // A frag row within 16-row subtile
    const int kp     = (lane >> 4) * 2;    // A frag: lanes<16 -> K{0,1}, lanes>=16 -> K{2,3}
    const int klo    = lane >> 4;          // B frag vgpr0 K index; vgpr1 is +2
    const int ncol   = lane & 15;

    for (int kc = 0; kc < C_; kc += KCHUNK) {
        // cooperative weight-tile load: coalesced global reads (contiguous kk),
        // padded LDS writes (stride 17 -> conflict free)
        for (int idx = threadIdx.x; idx < KCHUNK * 16; idx += 256) {
            const int kk = idx & (KCHUNK - 1);
            const int n  = idx >> 8;
            const long go = (long)(nBase + n) * C_ + kc + kk;
            ldsW[0 * KCHUNK * LPITCH + kk * LPITCH + n] = Wk[go];
            ldsW[1 * KCHUNK * LPITCH + kk * LPITCH + n] = Wv[go];
            ldsW[2 * KCHUNK * LPITCH + kk * LPITCH + n] = Wr[go];
        }
        __syncthreads();

        #pragma unroll 2
        for (int k0 = 0; k0 < KCHUNK; k0 += 4) {
            v2f a[MSUB];
            #pragma unroll
            for (int s = 0; s < MSUB; ++s) {
                const float* xp =
                    x + (long)(mBase + s * 16 + rowOff) * C_ + kc + k0 + kp;
                a[s].x = xp[0];
                a[s].y = xp[1];
            }

            const int bi = (k0 + klo) * LPITCH + ncol;
            v2f bK, bV, bRf;
            bK.x  = ldsW[0 * KCHUNK * LPITCH + bi];
            bK.y  = ldsW[0 * KCHUNK * LPITCH + bi + 2 * LPITCH];
            bV.x  = ldsW[1 * KCHUNK * LPITCH + bi];
            bV.y  = ldsW[1 * KCHUNK * LPITCH + bi + 2 * LPITCH];
            bRf.x = ldsW[2 * KCHUNK * LPITCH + bi];
            bRf.y = ldsW[2 * KCHUNK * LPITCH + bi + 2 * LPITCH];

            #pragma unroll
            for (int s = 0; s < MSUB; ++s) {
                accK[s] = __builtin_amdgcn_wmma_f32_16x16x4_f32(
                    false, a[s], false, bK, (short)0, accK[s], false, false);
                accV[s] = __builtin_amdgcn_wmma_f32_16x16x4_f32(
                    false, a[s], false, bV, (short)0, accV[s], false, false);
                accR[s] = __builtin_amdgcn_wmma_f32_16x16x4_f32(
                    false, a[s], false, bRf, (short)0, accR[s], false, false);
            }
        }
        __syncthreads();
    }

    // C/D layout: VGPR r -> M = r (lanes 0-15) / M = 8+r (lanes 16-31); N = lane&15
    #pragma unroll
    for (int s = 0; s < MSUB; ++s) {
        const int mOff = mBase + s * 16 + 8 * (lane >> 4);
        #pragma unroll
        for (int r2 = 0; r2 < 8; ++r2) {
            const long o = (long)(mOff + r2) * C_ + nBase + ncol;
            k[o]  = accK[s][r2];
            v[o]  = accV[s][r2];
            sr[o] = 1.0f / (1.0f + expf(-accR[s][r2]));
        }
    }
}

// ---------------------------------------------------------------------------
// Bidirectional stabilized WKV scan. One thread per (b, c) channel column.
// Backward pass stores exclusive suffix state (aR,bR,pR); forward pass keeps
// exclusive prefix state in registers and combines. ys may alias aR.
// ---------------------------------------------------------------------------
__global__ __launch_bounds__(256) void scan_kernel(
    const float* __restrict__ k, const float* __restrict__ v,
    const float* __restrict__ sr, const float* __restrict__ decay,
    const float* __restrict__ first,
    float* __restrict__ aR, float* __restrict__ bR, float* __restrict__ pR,
    float* __restrict__ ys)
{
    const int g = blockIdx.x * blockDim.x + threadIdx.x;
    if (g >= B_ * C_) return;
    const int c = g % C_;
    const int b = g / C_;
    const float w = decay[c] / (float)T_;
    const float u = first[c] / (float)T_;
    const long base = (long)b * T_ * C_ + c;
    const float NEG = -1e38f;

    // backward (suffix) pass: emit carry BEFORE update -> exclusive suffix
    float a = 0.f, bb = 0.f, p = NEG;
    for (int t = T_ - 1; t >= 0; --t) {
        const long o = base + (long)t * C_;
        aR[o] = a; bR[o] = bb; pR[o] = p;
        const float kt = k[o], vt = v[o];
        const float pn = fmaxf(p - w, kt);
        const float e1 = expf(p - w - pn);
        const float e2 = expf(kt - pn);
        a  = e1 * a + e2 * vt;
        bb = e1 * bb + e2;
        p  = pn;
    }

    // forward (prefix) pass + combine
    a = 0.f; bb = 0.f; p = NEG;
    for (int t = 0; t < T_; ++t) {
        const long o = base + (long)t * C_;
        const float kt = k[o], vt = v[o];
        const float ar = aR[o], br = bR[o], pr = pR[o];
        const float ps = u + kt;
        const float q  = fmaxf(fmaxf(p, pr), ps);
        const float eL = expf(p - q);
        const float eR = expf(pr - q);
        const float eS = expf(ps - q);
        const float y  = (eL * a + eR * ar + eS * vt) / (eL * bb + eR * br + eS);
        ys[o] = sr[o] * y;                       // overwrite aR slot already consumed
        const float pn = fmaxf(p - w, kt);
        const float e1 = expf(p - w - pn);
        const float e2 = expf(kt - pn);
        a  = e1 * a + e2 * vt;
        bb = e1 * bb + e2;
        p  = pn;
    }
}

// ---------------------------------------------------------------------------
// Output projection: out = ys * Wo^T  (out[b,t,c] = sum_j ys[b,t,j]*Wo[c,j])
// Same 64x16 wave-tile scheme, single matrix.
// ---------------------------------------------------------------------------
__global__ __launch_bounds__(256) void out_kernel(
    const float* __restrict__ ys, const float* __restrict__ Wo,
    float* __restrict__ out)
{
    __shared__ float ldsW[KCHUNK * LPITCH];

    const int nTile  = blockIdx.x % NTILES;
    const int mBlock = blockIdx.x / NTILES;
    const int wave   = threadIdx.x >> 5;
    const int lane   = threadIdx.x & 31;
    const int mBase  = mBlock * (8 * 16 * MSUB) + wave * (16 * MSUB);
    const int nBase  = nTile * 16;

    v8f acc[MSUB] = {};
    const int rowOff = lane & 15;
    const int kp     = (lane >> 4) * 2;
    const int klo    = lane >> 4;
    const int ncol   = lane & 15;

    for (int kc = 0; kc < C_; kc += KCHUNK) {
        for (int idx = threadIdx.x; idx < KCHUNK * 16; idx += 256) {
            const int kk = idx & (KCHUNK - 1);
            const int n  = idx >> 8;
            ldsW[kk * LPITCH + n] = Wo[(long)(nBase + n) * C_ + kc + kk];
        }
        __syncthreads();

        #pragma unroll 2
        for (int k0 = 0; k0 < KCHUNK; k0 += 4) {
            v2f a[MSUB];
            #pragma unroll
            for (int s = 0; s < MSUB; ++s) {
                const float* yp =
                    ys + (long)(mBase + s * 16 + rowOff) * C_ + kc + k0 + kp;
                a[s].x = yp[0];
                a[s].y = yp[1];
            }
            const int bi = (k0 + klo) * LPITCH + ncol;
            v2f bW;
            bW.x = ldsW[bi];
            bW.y = ldsW[bi + 2 * LPITCH];
            #pragma unroll
            for (int s = 0; s < MSUB; ++s) {
                acc[s] = __builtin_amdgcn_wmma_f32_16x16x4_f32(
                    false, a[s], false, bW, (short)0, acc[s], false, false);
            }
        }
        __syncthreads();
    }

    #pragma unroll
    for (int s = 0; s < MSUB; ++s) {
        const int mOff = mBase + s * 16 + 8 * (lane >> 4);
        #pragma unroll
        for (int r2 = 0; r2 < 8; ++r2) {
            out[(long)(mOff + r2) * C_ + nBase + ncol] = acc[s][r2];
        }
    }
}

// ---------------------------------------------------------------------------
extern "C" void kernel_launch(void* const* d_in, const int* in_sizes, int n_in,
                              void* d_out, int out_size, void* d_ws, size_t ws_size,
                              hipStream_t stream)
{
    const float* x     = (const float*)d_in[0];
    const float* Wk    = (const float*)d_in[1];
    const float* Wv    = (const float*)d_in[2];
    const float* Wr    = (const float*)d_in[3];
    const float* Wo    = (const float*)d_in[4];
    const float* decay = (const float*)d_in[5];
    const float* first = (const float*)d_in[6];
    float* out = (float*)d_out;

    const size_t S = (size_t)M_ * C_;            // 12.58M floats per buffer
    float* k   = (float*)d_ws;
    float* v   = k  + S;
    float* sr  = v  + S;
    float* aR  = sr + S;
    float* bRb = aR + S;
    float* pRb = bRb + S;
    float* ys  = aR;                             // reuse aR buffer for sr*y

    dim3 blk(256);
    dim3 gGemm(NTILES * (M_ / (8 * 16 * MSUB))); // 48 * 32 = 1536 blocks
    proj_kernel<<<gGemm, blk, 0, stream>>>(x, Wk, Wv, Wr, k, v, sr);

    scan_kernel<<<dim3((B_ * C_ + 255) / 256), blk, 0, stream>>>(
        k, v, sr, decay, first, aR, bRb, pRb, ys);

    out_kernel<<<gGemm, blk, 0, stream>>>(ys, Wo, out);
}